// RegionLoss_65549790871854
// MI455X (gfx1250) — compile-verified
//
#include <hip/hip_runtime.h>
#include <math.h>

#define NA 5
#define NC 20
#define NH 38
#define NW 38
#define CPA (NH * NW)          // 1444 cells per anchor-plane
#define CPB (NA * CPA)         // 7220 cells per batch
#define MAXB 50
#define SIL 0.6f
#define OBJ_SCALE 5.0f
#define REC 16                 // floats per gt record
#define BLK 256
#define BLOCKS_PER_B ((CPB + BLK - 1) / BLK)   // 29

__constant__ float c_aw[NA] = {1.3221f, 3.19275f, 5.05587f, 9.47112f, 11.2364f};
__constant__ float c_ah[NA] = {1.73145f, 4.00944f, 8.09892f, 4.84053f, 10.0071f};

typedef float v2f __attribute__((ext_vector_type(2)));
typedef float v8f __attribute__((ext_vector_type(8)));

__device__ __forceinline__ float sigmoidf_(float v) { return 1.0f / (1.0f + expf(-v)); }

__device__ __forceinline__ float iou_cwh_(float x1, float y1, float w1, float h1,
                                          float x2, float y2, float w2, float h2) {
  float uw = fmaxf(x1 + 0.5f * w1, x2 + 0.5f * w2) - fminf(x1 - 0.5f * w1, x2 - 0.5f * w2);
  float uh = fmaxf(y1 + 0.5f * h1, y2 + 0.5f * h2) - fminf(y1 - 0.5f * h1, y2 - 0.5f * h2);
  float cw = w1 + w2 - uw;
  float ch = h1 + h2 - uh;
  float inter = (cw <= 0.0f || ch <= 0.0f) ? 0.0f : cw * ch;
  return inter / (w1 * h1 + w2 * h2 - inter);
}

// Exact f32 wave32 sum via the matrix pipe: D = A x ones + 0 row-sums A.
// A 16x4 layout: lanes 0-15 give A[m][0],A[m][1]; lanes 16-31 give A[m][2],A[m][3].
// Put per-lane value in A[0] -> D[m][n] = v(lane m) + v(lane m+16), constant over n.
// Per-lane sum of its 8 D regs = sum(rows 0-7) on lanes 0-15, sum(rows 8-15) on 16-31.
__device__ __forceinline__ float wave_sum_wmma(float v) {
  v2f A; A[0] = v;     A[1] = 0.0f;
  v2f B; B[0] = 1.0f;  B[1] = 1.0f;
  v8f C = {};
  C = __builtin_amdgcn_wmma_f32_16x16x4_f32(false, A, false, B, (short)0, C, false, false);
  float s = C[0] + C[1] + C[2] + C[3] + C[4] + C[5] + C[6] + C[7];
  s += __shfl_xor(s, 16, 32);
  return s;  // full wave total, all lanes
}

// ---------------- K1: per-(b,t) ground-truth records ----------------
// rec[(b*MAXB+t)*REC + f]: 0..3 gx,gy,gw,gh | 4 valid | 5 best_n | 6 gi | 7 gj
//                          8..11 tx,ty,tw,th | 12 iou_gt | 13 gcls
__global__ void gt_kernel(const float* __restrict__ out, const float* __restrict__ tgt,
                          float* __restrict__ rec) {
  int b = blockIdx.x;
  int t = threadIdx.x;
  __shared__ float sx[MAXB];
  if (t < MAXB) sx[t] = tgt[(size_t)b * MAXB * 5 + t * 5 + 1];
  __syncthreads();
  if (t >= MAXB) return;

  const float* tg = tgt + (size_t)b * MAXB * 5 + t * 5;
  float gcls = tg[0];
  float gx = tg[1] * NW, gy = tg[2] * NH, gw = tg[3] * NW, gh = tg[4] * NH;

  int valid = 1;
  for (int s = 0; s <= t; ++s) valid &= (sx[s] != 0.0f);

  // best anchor by centered wh-IoU (first max wins, like argmax)
  int bn = 0; float best = -1.0f;
  for (int a = 0; a < NA; ++a) {
    float inter = fminf(gw, c_aw[a]) * fminf(gh, c_ah[a]);
    float iou = inter / (gw * gh + c_aw[a] * c_ah[a] - inter);
    if (iou > best) { best = iou; bn = a; }
  }
  int gi = min(max((int)gx, 0), NW - 1);
  int gj = min(max((int)gy, 0), NH - 1);

  size_t base = ((size_t)b * (NA * (5 + NC)) + (size_t)bn * (5 + NC)) * CPA +
                (size_t)gj * NW + gi;
  float o0 = out[base];
  float o1 = out[base + 1 * CPA];
  float o2 = out[base + 2 * CPA];
  float o3 = out[base + 3 * CPA];
  float px = sigmoidf_(o0) + (float)gi;
  float py = sigmoidf_(o1) + (float)gj;
  float pw = expf(o2) * c_aw[bn];
  float ph = expf(o3) * c_ah[bn];
  float iou_gt = iou_cwh_(gx, gy, gw, gh, px, py, pw, ph);

  float* r = rec + ((size_t)b * MAXB + t) * REC;
  r[0] = gx; r[1] = gy; r[2] = gw; r[3] = gh;
  r[4] = (float)valid; r[5] = (float)bn; r[6] = (float)gi; r[7] = (float)gj;
  r[8] = gx - (float)gi; r[9] = gy - (float)gj;
  r[10] = logf(gw / c_aw[bn]); r[11] = logf(gh / c_ah[bn]);
  r[12] = iou_gt; r[13] = gcls; r[14] = 0.0f; r[15] = 0.0f;
}

// ---------------- K2: init winner array ----------------
__global__ void init_winner(int* __restrict__ winner, int n) {
  int i = blockIdx.x * blockDim.x + threadIdx.x;
  if (i < n) winner[i] = -1;
}

// ---------------- K3: "last valid t wins" == atomicMax over t (deterministic) --------
__global__ void scatter_winner(const float* __restrict__ rec, int* __restrict__ winner) {
  int b = blockIdx.x;
  int t = threadIdx.x;
  if (t >= MAXB) return;
  const float* r = rec + ((size_t)b * MAXB + t) * REC;
  if (r[4] == 0.0f) return;
  int bn = (int)r[5], gi = (int)r[6], gj = (int)r[7];
  size_t cell = (size_t)b * CPB + (size_t)bn * CPA + (size_t)gj * NW + gi;
  atomicMax(&winner[cell], t);
}

// ---------------- K4: main streaming loss pass ----------------
__global__ void main_kernel(const float* __restrict__ out, const float* __restrict__ rec,
                            const int* __restrict__ winner, float* __restrict__ partials) {
  int b   = blockIdx.x / BLOCKS_PER_B;
  int blk = blockIdx.x % BLOCKS_PER_B;
  int tid = threadIdx.x;

  __shared__ float s_box[MAXB][5];  // gx,gy,gw,gh,valid for this batch
  for (int k = tid; k < MAXB * 5; k += BLK) {
    int t = k / 5, f = k % 5;
    s_box[t][f] = rec[((size_t)b * MAXB + t) * REC + f];
  }
  __syncthreads();

  int local = blk * BLK + tid;
  float loss = 0.0f;
  if (local < CPB) {
    int a = local / CPA;
    int remn = local % CPA;
    int j = remn / NW, i = remn % NW;
    size_t base = ((size_t)b * (NA * (5 + NC)) + (size_t)a * (5 + NC)) * CPA + remn;

    float o0 = out[base];
    float o1 = out[base + 1 * CPA];
    float o2 = out[base + 2 * CPA];
    float o3 = out[base + 3 * CPA];
    float o4 = out[base + 4 * CPA];

    // winner first: lets us prefetch every 4th of the 20 strided logit lines (rare positive
    // cells) and hide that latency behind the 50-iteration IoU loop below.
    int wt = winner[(size_t)b * CPB + local];
    if (wt >= 0) {
#pragma unroll
      for (int c = 0; c < NC; c += 4)
        __builtin_prefetch(out + base + (size_t)(5 + c) * CPA, 0, 1);
    }

    float x    = sigmoidf_(o0);
    float y    = sigmoidf_(o1);
    float conf = sigmoidf_(o4);
    float px = x + (float)i;
    float py = y + (float)j;
    float pw = expf(o2) * c_aw[a];
    float ph = expf(o3) * c_ah[a];

    float mx = 0.0f;
#pragma unroll 5
    for (int t = 0; t < MAXB; ++t) {
      if (s_box[t][4] != 0.0f) {
        float iou = iou_cwh_(px, py, pw, ph,
                             s_box[t][0], s_box[t][1], s_box[t][2], s_box[t][3]);
        mx = fmaxf(mx, iou);
      }
    }

    float conf_mask = (mx > SIL) ? 0.0f : 1.0f;
    float txc = 0.5f, tyc = 0.5f, twc = 0.0f, thc = 0.0f, tcf = 0.0f;
    float clsm = 0.0f; int lbl = 0;
    if (wt >= 0) {
      const float* r = rec + ((size_t)b * MAXB + wt) * REC;
      conf_mask = OBJ_SCALE; clsm = 1.0f;
      txc = r[8]; tyc = r[9]; twc = r[10]; thc = r[11];
      tcf = r[12]; lbl = (int)r[13];
    }

    float dx = x - txc, dy = y - tyc, dw = o2 - twc, dh = o3 - thc;
    loss = 0.5f * (dx * dx + dy * dy + dw * dw + dh * dh);
    float cmf = sqrtf(conf_mask);
    float dc = conf * cmf - tcf * cmf;
    loss += 0.5f * dc * dc;

    if (clsm != 0.0f) {
      float lg[NC]; float m = -INFINITY;
#pragma unroll
      for (int c = 0; c < NC; ++c) {
        lg[c] = out[base + (size_t)(5 + c) * CPA];
        m = fmaxf(m, lg[c]);
      }
      float se = 0.0f;
#pragma unroll
      for (int c = 0; c < NC; ++c) se += expf(lg[c] - m);
      float lse = m + logf(se);
      loss += (lse - lg[lbl]);  // CLASS_SCALE = 1
    }
  }

  // all 256 threads reach here (full waves -> EXEC all ones for WMMA)
  float s = wave_sum_wmma(loss);
  __shared__ float swave[BLK / 32];
  if ((tid & 31) == 0) swave[tid >> 5] = s;
  __syncthreads();
  if (tid == 0) {
    float tot = 0.0f;
#pragma unroll
    for (int k = 0; k < BLK / 32; ++k) tot += swave[k];
    partials[blockIdx.x] = tot;
  }
}

// ---------------- K5: fixed-order final reduction (deterministic) ----------------
__global__ void reduce_kernel(const float* __restrict__ partials, int n,
                              float* __restrict__ outv) {
  int tid = threadIdx.x;
  float acc = 0.0f;
  for (int k = tid; k < n; k += BLK) acc += partials[k];
  float s = wave_sum_wmma(acc);
  __shared__ float swave[BLK / 32];
  if ((tid & 31) == 0) swave[tid >> 5] = s;
  __syncthreads();
  if (tid == 0) {
    float tot = 0.0f;
#pragma unroll
    for (int k = 0; k < BLK / 32; ++k) tot += swave[k];
    outv[0] = tot;
  }
}

extern "C" void kernel_launch(void* const* d_in, const int* in_sizes, int n_in,
                              void* d_out, int out_size, void* d_ws, size_t ws_size,
                              hipStream_t stream) {
  (void)n_in; (void)out_size; (void)ws_size;
  const float* d_output = (const float*)d_in[0];
  const float* d_target = (const float*)d_in[1];
  int nB = in_sizes[0] / (NA * (5 + NC) * NH * NW);  // 64

  float* rec = (float*)d_ws;                        // nB*50*16 floats
  size_t recN = (size_t)nB * MAXB * REC;
  int* winner = (int*)(rec + recN);                 // nB*7220 ints
  size_t winN = (size_t)nB * CPB;
  float* partials = (float*)(winner + winN);        // nB*29 floats

  int nBlocks = nB * BLOCKS_PER_B;

  gt_kernel<<<dim3(nB), dim3(64), 0, stream>>>(d_output, d_target, rec);
  init_winner<<<dim3((int)((winN + 255) / 256)), dim3(256), 0, stream>>>(winner, (int)winN);
  scatter_winner<<<dim3(nB), dim3(64), 0, stream>>>(rec, winner);
  main_kernel<<<dim3(nBlocks), dim3(BLK), 0, stream>>>(d_output, rec, winner, partials);
  reduce_kernel<<<dim3(1), dim3(BLK), 0, stream>>>(partials, nBlocks, (float*)d_out);
}